// UNetModel100k_quan_66975720014022
// MI455X (gfx1250) — compile-verified
//
#include <hip/hip_runtime.h>
#include <math.h>

// ---------------------------------------------------------------------------
// Quantized depthwise-separable UNet forward, fused for MI455X (gfx1250).
// BW-bound workload (~2+ GB activation traffic vs ~18 GFLOP) -> fp32 exact
// path with V_WMMA_F32_16X16X4_F32 for all pointwise GEMMs.
//   - templated (CI_PAD, CO): K-loop fully unrolled, guard-free (padded
//     weights/bias/LDS), single b64 loads per operand pair
//   - dual accumulators per wave: shared B operand, independent WMMA chains
//   - 32-bit indexing + interior fast path for the depthwise phase
//   - upsample + concat folded into fused-kernel addressing
//   - qleaky per-tensor scale: block abs-max + atomicMax(u32) + requant pass
// ---------------------------------------------------------------------------

typedef __attribute__((ext_vector_type(2))) float v2f;
typedef __attribute__((ext_vector_type(8))) float v8f;

#define QMAX 32767.0f           /* BIT = 16 */
#define NSLOPE 0.1015625f
#define PITCH 160               /* floats per k-pair LDS row: 160 mod 64 == 32
                                   -> half-waves hit disjoint bank halves     */

// ---- depthwise-weight fake-quant (unpadded, [c][9]) ------------------------
__global__ void __launch_bounds__(256) wquant_kernel(const float* __restrict__ src,
                                                     float* __restrict__ dst, int n) {
  __shared__ float red[256];
  float m = 0.f;
  for (int i = threadIdx.x; i < n; i += 256) m = fmaxf(m, fabsf(src[i]));
  red[threadIdx.x] = m;
  __syncthreads();
  for (int s = 128; s > 0; s >>= 1) {
    if (threadIdx.x < s) red[threadIdx.x] = fmaxf(red[threadIdx.x], red[threadIdx.x + s]);
    __syncthreads();
  }
  const float scale = fmaxf(red[0] / QMAX, 1e-8f);
  for (int i = threadIdx.x; i < n; i += 256) {
    float q = rintf(src[i] / scale);            // round-half-even == jnp.round
    q = fminf(fmaxf(q, -QMAX), QMAX);
    dst[i] = q * scale;
  }
}

// ---- pointwise-weight fake-quant into padded [coPad][cip] ------------------
__global__ void __launch_bounds__(256) wquant_pad_kernel(const float* __restrict__ src,
                                                         float* __restrict__ dst,
                                                         int co, int ci, int coPad, int cip) {
  __shared__ float red[256];
  const int n = co * ci;
  float m = 0.f;
  for (int i = threadIdx.x; i < n; i += 256) m = fmaxf(m, fabsf(src[i]));
  red[threadIdx.x] = m;
  __syncthreads();
  for (int s = 128; s > 0; s >>= 1) {
    if (threadIdx.x < s) red[threadIdx.x] = fmaxf(red[threadIdx.x], red[threadIdx.x + s]);
    __syncthreads();
  }
  const float scale = fmaxf(red[0] / QMAX, 1e-8f);
  const int tot = coPad * cip;
  for (int i = threadIdx.x; i < tot; i += 256) {
    const int r = i / cip, k = i % cip;
    float v = 0.f;
    if (r < co && k < ci) {
      float q = rintf(src[r * ci + k] / scale);
      v = fminf(fmaxf(q, -QMAX), QMAX) * scale;
    }
    dst[i] = v;
  }
}

__global__ void biaspad_kernel(const float* __restrict__ src, float* __restrict__ dst,
                               int co, int coPad) {
  const int i = threadIdx.x;
  if (i < coPad) dst[i] = (i < co) ? src[i] : 0.f;
}

__global__ void zero_kernel(unsigned int* p, int n) {
  const int i = blockIdx.x * blockDim.x + threadIdx.x;
  if (i < n) p[i] = 0u;
}

// ---- fused depthwise3x3 + pointwise(WMMA) + bias + leaky + absmax ----------
// Workgroup tile: 16 wide x 4 high = 64 output pixels of one image.
// in0: c0 channels (optionally x2-nearest-upsampled on read); in1: c1 skip.
// LDS layout: sdw[(k>>1)*PITCH + p*2 + (k&1)]  -> B operand is one ds_load_b64.
template <int CIP, int CO, bool RELU>
__global__ void __launch_bounds__(256) dsfused_t(
    const float* __restrict__ in0, int c0, int up0,
    const float* __restrict__ in1, int c1,
    int H, int W,
    const float* __restrict__ dwq,    // [ci][9] quantized
    const float* __restrict__ pwq,    // [coPad][CIP] quantized, zero-padded
    const float* __restrict__ biasp,  // [coPad] zero-padded
    float* __restrict__ out, unsigned int* __restrict__ amax) {
  constexpr int COP = ((CO + 15) / 16) * 16;
  constexpr int coT = COP / 16;
  constexpr int nT  = 4 * coT;        // 4 pixel tiles x co tiles
  __shared__ float sdw[(CIP / 2) * PITCH];
  __shared__ unsigned int blkmax;

  const int ci = c0 + c1;             // actual ci (== CIP except block 0: 3)
  const int tilesW = W >> 4;
  const int tilesH = H >> 2;
  int bid = blockIdx.x;
  const int tx = bid % tilesW;  bid /= tilesW;
  const int ty = bid % tilesH;  bid /= tilesH;
  const int n  = bid;
  const int x0 = tx << 4, y0 = ty << 2;
  const int H0 = up0 ? (H >> 1) : H, W0 = up0 ? (W >> 1) : W;
  // block-uniform: tile + 1-px halo fully inside the image?
  const bool interior = (x0 >= 1) & (y0 >= 1) & (x0 + 17 <= W) & (y0 + 5 <= H);

  // ---- phase 1: depthwise 3x3 SAME -> LDS (upsample/concat fused in) ----
  for (int idx = threadIdx.x; idx < CIP * 64; idx += 256) {
    const int c  = idx >> 6;
    const int p  = idx & 63;
    float acc = 0.f;
    if (c < ci) {
      const int px = p & 15, py = p >> 4;
      const int y = y0 + py, x = x0 + px;
      const float* src; int cc, cs, Ws, ups;
      int HsWs;
      if (c < c0) { src = in0; cc = c;      cs = c0; Ws = W0; ups = up0; HsWs = H0 * W0; }
      else        { src = in1; cc = c - c0; cs = c1; Ws = W;  ups = 0;   HsWs = H * W;   }
      const unsigned sbase = ((unsigned)n * cs + cc) * (unsigned)HsWs;
      const float* wp = dwq + c * 9;
      if (interior) {                      // unguarded 9-tap fast path
#pragma unroll
        for (int ky = 0; ky < 3; ++ky) {
          const int yy = y + ky - 1;
          const unsigned rowb = sbase + (unsigned)(ups ? (yy >> 1) : yy) * Ws;
#pragma unroll
          for (int kx = 0; kx < 3; ++kx) {
            const int xx = x + kx - 1;
            const float v = src[rowb + (unsigned)(ups ? (xx >> 1) : xx)];
            acc = fmaf(wp[ky * 3 + kx], v, acc);
          }
        }
      } else {                             // border tiles: guarded taps
#pragma unroll
        for (int ky = 0; ky < 3; ++ky) {
          const int yy = y + ky - 1;
          const unsigned rowb = sbase + (unsigned)(ups ? (yy >> 1) : yy) * Ws;
#pragma unroll
          for (int kx = 0; kx < 3; ++kx) {
            const int xx = x + kx - 1;
            float v = 0.f;
            if (yy >= 0 && yy < H && xx >= 0 && xx < W)
              v = src[rowb + (unsigned)(ups ? (xx >> 1) : xx)];
            acc = fmaf(wp[ky * 3 + kx], v, acc);
          }
        }
      }
    }
    sdw[(c >> 1) * PITCH + (p << 1) + (c & 1)] = acc;
  }
  __syncthreads();
  if (threadIdx.x == 0) blkmax = 0u;

  // ---- phase 2: pointwise GEMM via V_WMMA_F32_16X16X4_F32 ----
  const int lane = threadIdx.x & 31;
  const int wave = threadIdx.x >> 5;
  const int nidx = lane & 15;                 // N coord (pixel within row)
  const int hi   = lane >> 4;                 // half-wave: carries K, K+2
  const unsigned HW = (unsigned)H * (unsigned)W;
  float lmax = 0.f;

  for (int t = wave; t < nT; t += 16) {       // wave-uniform: EXEC stays full
    const int pt  = t & 3;                    // pixel tile == output row
    const int ct0 = t >> 2;
    const int ct1 = ct0 + 2;                  // (t+8) has same pt -> shares B
    const bool dual = (t + 8) < nT;           // wave-uniform
    const int pixoff = (pt << 4) + nidx;
    const v2f* bb = (const v2f*)sdw + pixoff; // row stride PITCH/2 v2f
    const v2f* a0 = (const v2f*)(pwq + (size_t)((ct0 << 4) + nidx) * CIP);
    const v2f* a1 = (const v2f*)(pwq + (size_t)((ct1 << 4) + nidx) * CIP);
    const int m0 = (ct0 << 4) + (hi << 3);
    const int m1 = (ct1 << 4) + (hi << 3);

    v8f acc0, acc1;
#pragma unroll
    for (int v = 0; v < 8; ++v) {
      acc0[v] = biasp[m0 + v];
      acc1[v] = dual ? biasp[m1 + v] : 0.f;
    }

    if (dual) {
#pragma unroll
      for (int k2 = 0; k2 < CIP / 2; k2 += 2) {     // k2 = K>>1, step K=4
        const int r = k2 + hi;
        const v2f b   = bb[r * (PITCH / 2)];
        const v2f av0 = a0[r];
        const v2f av1 = a1[r];
        acc0 = __builtin_amdgcn_wmma_f32_16x16x4_f32(false, av0, false, b,
                                                     (short)0, acc0, false, false);
        acc1 = __builtin_amdgcn_wmma_f32_16x16x4_f32(false, av1, false, b,
                                                     (short)0, acc1, false, false);
      }
    } else {
#pragma unroll
      for (int k2 = 0; k2 < CIP / 2; k2 += 2) {
        const int r = k2 + hi;
        const v2f b   = bb[r * (PITCH / 2)];
        const v2f av0 = a0[r];
        acc0 = __builtin_amdgcn_wmma_f32_16x16x4_f32(false, av0, false, b,
                                                     (short)0, acc0, false, false);
      }
    }

    // ---- epilogue: 32-bit addressing; consecutive channels differ by HW ----
    const int y = y0 + pt, x = x0 + nidx;
    const unsigned base0 =
        (((unsigned)n * (unsigned)CO + (unsigned)m0) * (unsigned)H + (unsigned)y)
            * (unsigned)W + (unsigned)x;
#pragma unroll
    for (int v = 0; v < 8; ++v) {
      float z = acc0[v];
      if (RELU) z = (z >= 0.f) ? z : NSLOPE * z;
      if constexpr (CO % 16 == 0) {
        out[base0 + (unsigned)v * HW] = z;
        lmax = fmaxf(lmax, fabsf(z));
      } else {
        if (m0 + v < CO) {
          out[base0 + (unsigned)v * HW] = z;
          lmax = fmaxf(lmax, fabsf(z));
        }
      }
    }
    if (dual) {
      const unsigned base1 = base0 + (unsigned)(m1 - m0) * HW;
#pragma unroll
      for (int v = 0; v < 8; ++v) {
        float z = acc1[v];
        if (RELU) z = (z >= 0.f) ? z : NSLOPE * z;
        if constexpr (CO % 16 == 0) {
          out[base1 + (unsigned)v * HW] = z;
          lmax = fmaxf(lmax, fabsf(z));
        } else {
          if (m1 + v < CO) {
            out[base1 + (unsigned)v * HW] = z;
            lmax = fmaxf(lmax, fabsf(z));
          }
        }
      }
    }
  }

  if (RELU) {                               // per-tensor abs-max for qleaky
    __syncthreads();
    atomicMax(&blkmax, __float_as_uint(lmax));
    __syncthreads();
    if (threadIdx.x == 0) atomicMax(amax, blkmax);
  }
}

// ---- elementwise activation fake-quant (second pass of qleaky) -------------
__global__ void __launch_bounds__(256) aquant_kernel(float* __restrict__ buf,
                                                     const unsigned int* __restrict__ amax,
                                                     long n) {
  const float s = fmaxf(__uint_as_float(*amax) / QMAX, 1e-8f);
  const float inv = 1.0f / s;
  for (long i = (long)blockIdx.x * blockDim.x + threadIdx.x; i < n;
       i += (long)gridDim.x * blockDim.x) {
    float q = rintf(buf[i] * inv);
    q = fminf(fmaxf(q, -QMAX), QMAX);
    buf[i] = q * s;
  }
}

// ---- 2x2 max pool ----------------------------------------------------------
__global__ void __launch_bounds__(256) pool_kernel(const float* __restrict__ in,
                                                   float* __restrict__ out,
                                                   int C, int Ho, int Wo, int N) {
  const unsigned i = blockIdx.x * 256u + threadIdx.x;
  const unsigned total = (unsigned)N * C * Ho * Wo;
  if (i >= total) return;
  const unsigned x = i % (unsigned)Wo; unsigned r = i / (unsigned)Wo;
  const unsigned y = r % (unsigned)Ho; r /= (unsigned)Ho;
  const unsigned Wi = (unsigned)Wo * 2;
  const float* p = in + (r * (unsigned)(Ho * 2) + y * 2) * Wi + x * 2;
  out[i] = fmaxf(fmaxf(p[0], p[1]), fmaxf(p[Wi], p[Wi + 1]));
}

// ---------------------------------------------------------------------------
static const int CI_[20] = {3,32,32,32,64,64,64,64,64,64,96,96,128,64,128,64,128,64,64,32};
static const int CO_[20] = {32,32,32,64,64,64,64,64,64,96,96,64,64,64,64,64,64,32,32,2};
static const int NB = 4;

template <int CIP, int CO, bool RELU>
static void launch_ds(const float* i0, int c0, int up0, const float* i1, int c1,
                      int H, int W, const float* dwq, const float* pwq,
                      const float* biasp, float* o, unsigned int* amax,
                      int co, hipStream_t stream) {
  dim3 grid((unsigned)(NB * (H / 4) * (W / 16)));
  dsfused_t<CIP, CO, RELU><<<grid, 256, 0, stream>>>(i0, c0, up0, i1, c1, H, W,
                                                     dwq, pwq, biasp, o, amax);
  if (RELU) {
    long n = (long)NB * co * H * W;
    long b = (n + 255) / 256; if (b > 4096) b = 4096;
    aquant_kernel<<<(unsigned)b, 256, 0, stream>>>(o, amax, n);
  }
}

extern "C" void kernel_launch(void* const* d_in, const int* in_sizes, int n_in,
                              void* d_out, int out_size, void* d_ws, size_t ws_size,
                              hipStream_t stream) {
  const int S = 512;
  (void)in_sizes; (void)n_in; (void)out_size; (void)ws_size;

  const float* x = (const float*)d_in[0];
  float* outp = (float*)d_out;

  // ---- padded shapes + workspace offsets ----
  int CIP[20], COP[20], qdwo[20], qpwo[20], biaso[20];
  { int a = 0, b = 0, c = 0;
    for (int i = 0; i < 20; ++i) {
      CIP[i] = (CI_[i] + 3) & ~3;
      COP[i] = ((CO_[i] + 15) / 16) * 16;
      qdwo[i] = a;  a += CI_[i] * 9;
      qpwo[i] = b;  b += COP[i] * CIP[i];
      biaso[i] = c; c += COP[i];
    } }

  float* ws = (float*)d_ws;
  unsigned int* scales = (unsigned int*)ws;            // 20 slots
  size_t off = 32;
  float* qdw  = ws + off; off += 16384;                // 12123 used
  float* qpw  = ws + off; off += 98304;                // padded pw, aligned
  float* bsb  = ws + off; off += 2048;                 // padded bias
  float* bufA = ws + off; off += (size_t)33554432;     // 4*32*512*512
  float* bufB = ws + off; off += (size_t)8388608;
  float* bufP = ws + off; off += (size_t)8388608;
  float* s1   = ws + off; off += (size_t)33554432;     // (N,32,512,512)
  float* s2   = ws + off; off += (size_t)16777216;     // (N,64,256,256)
  float* s3   = ws + off; off += (size_t)4194304;      // (N,64,128,128)
  float* s4   = ws + off; off += (size_t)1048576;      // (N,64,64,64)

  // ---- weight fake-quant (padded) + scale slot reset ----
  zero_kernel<<<1, 32, 0, stream>>>(scales, 20);
  for (int i = 0; i < 20; ++i) {
    wquant_kernel<<<1, 256, 0, stream>>>((const float*)d_in[1 + i],
                                         qdw + qdwo[i], CI_[i] * 9);
    wquant_pad_kernel<<<1, 256, 0, stream>>>((const float*)d_in[21 + i],
                                             qpw + qpwo[i], CO_[i], CI_[i],
                                             COP[i], CIP[i]);
    biaspad_kernel<<<1, COP[i], 0, stream>>>((const float*)d_in[41 + i],
                                             bsb + biaso[i], CO_[i], COP[i]);
  }

  auto run_pool = [&](const float* in, float* o, int C, int Ho, int Wo) {
    long tot = (long)NB * C * Ho * Wo;
    pool_kernel<<<(unsigned)((tot + 255) / 256), 256, 0, stream>>>(in, o, C, Ho, Wo, NB);
  };
#define DS(i, CIPc, COc, RELUc, i0, c0, up0, i1, c1, H, W, o)                   \
  launch_ds<CIPc, COc, RELUc>(i0, c0, up0, i1, c1, H, W, qdw + qdwo[i],         \
                              qpw + qpwo[i], bsb + biaso[i], o, scales + i,     \
                              CO_[i], stream)

  // ---- encoder ----
  DS(0,  4,   32, true, x,    3,  0, nullptr, 0, S,    S,    bufA);
  DS(1,  32,  32, true, bufA, 32, 0, nullptr, 0, S,    S,    s1);
  run_pool(s1, bufP, 32, S/2, S/2);
  DS(2,  32,  32, true, bufP, 32, 0, nullptr, 0, S/2,  S/2,  bufA);
  DS(3,  32,  64, true, bufA, 32, 0, nullptr, 0, S/2,  S/2,  s2);
  run_pool(s2, bufP, 64, S/4, S/4);
  DS(4,  64,  64, true, bufP, 64, 0, nullptr, 0, S/4,  S/4,  bufA);
  DS(5,  64,  64, true, bufA, 64, 0, nullptr, 0, S/4,  S/4,  s3);
  run_pool(s3, bufP, 64, S/8, S/8);
  DS(6,  64,  64, true, bufP, 64, 0, nullptr, 0, S/8,  S/8,  bufA);
  DS(7,  64,  64, true, bufA, 64, 0, nullptr, 0, S/8,  S/8,  s4);
  run_pool(s4, bufP, 64, S/16, S/16);
  DS(8,  64,  64, true, bufP, 64, 0, nullptr, 0, S/16, S/16, bufA);
  DS(9,  64,  96, true, bufA, 64, 0, nullptr, 0, S/16, S/16, bufB);
  // ---- dec_middle ----
  DS(10, 96,  96, true, bufB, 96, 0, nullptr, 0, S/16, S/16, bufA);
  DS(11, 96,  64, true, bufA, 96, 0, nullptr, 0, S/16, S/16, bufB);
  // ---- decoder (upsample + skip-concat fused into input addressing) ----
  DS(12, 128, 64, true, bufB, 64, 1, s4, 64, S/8,  S/8,  bufA);
  DS(13, 64,  64, true, bufA, 64, 0, nullptr, 0, S/8,  S/8,  bufB);
  DS(14, 128, 64, true, bufB, 64, 1, s3, 64, S/4,  S/4,  bufA);
  DS(15, 64,  64, true, bufA, 64, 0, nullptr, 0, S/4,  S/4,  bufB);
  DS(16, 128, 64, true, bufB, 64, 1, s2, 64, S/2,  S/2,  bufA);
  DS(17, 64,  32, true, bufA, 64, 0, nullptr, 0, S/2,  S/2,  bufB);
  DS(18, 64,  32, true, bufB, 32, 1, s1, 32, S,    S,    bufA);
  DS(19, 32,  2,  false, bufA, 32, 0, nullptr, 0, S,   S,    outp);  // no relu
#undef DS
}